// GNNHolographicPredictor_17136919511359
// MI455X (gfx1250) — compile-verified
//
#include <hip/hip_runtime.h>
#include <hip/hip_bf16.h>
#include <math.h>

#define NN      100000
#define EREAL   1600000
#define ETOT    (EREAL + NN)   // edges + self loops
#define HIDC    64

typedef __attribute__((ext_vector_type(2))) float v2f;
typedef __attribute__((ext_vector_type(8))) float v8f;

// ---------- ordered-float <-> uint for atomic max ----------
__device__ __forceinline__ unsigned ordered_u(float f) {
  unsigned u = __float_as_uint(f);
  return (u & 0x80000000u) ? ~u : (u | 0x80000000u);
}
__device__ __forceinline__ float ordered_f(unsigned u) {
  return (u & 0x80000000u) ? __uint_as_float(u & 0x7FFFFFFFu)
                           : __uint_as_float(~u);
}

// ---------- degree / norm ----------
__global__ void k_init_deg(float* deg) {
  int i = blockIdx.x * blockDim.x + threadIdx.x;
  if (i < NN) deg[i] = 1.0f;                  // self-loop contribution
}
__global__ void k_degree(const int* __restrict__ dst, float* deg) {
  int e = blockIdx.x * blockDim.x + threadIdx.x;
  if (e < EREAL) atomicAdd(&deg[dst[e]], 1.0f);
}
__global__ void k_rsqrt(float* deg) {
  int i = blockIdx.x * blockDim.x + threadIdx.x;
  if (i < NN) deg[i] = rsqrtf(deg[i]);        // deg >= 1 always
}

// ---------- WMMA fp32 GEMM:  Y[NN,64] = X[NN,K] @ W[K,64] ----------
// One wave per 16-row tile; 4 column tiles; K/4 k-steps of v_wmma_f32_16x16x4_f32.
// W staged once per block into LDS, k-pair interleaved so each B fragment
// {W[k0][n], W[k0+1][n]} is one ds_load_b64.
template <int K>
__global__ void k_gemm_wmma(const float* __restrict__ X, const float* __restrict__ W,
                            float* __restrict__ Y) {
  __shared__ float ldsW[(K / 2) * 128];       // K*64 floats, pair-interleaved

  // cooperative stage: lds[(k>>1)*128 + n*2 + (k&1)] = W[k*64 + n]
  for (int i = threadIdx.x; i < K * HIDC; i += blockDim.x) {
    int k = i >> 6;
    int n = i & 63;
    ldsW[(k >> 1) * 128 + n * 2 + (k & 1)] = W[i];
  }
  __syncthreads();

  int wave = (int)((blockIdx.x * blockDim.x + threadIdx.x) >> 5);
  int lane = threadIdx.x & 31;
  const int ntiles = NN / 16;                 // 6250, exact
  if (wave >= ntiles) return;                 // wave-uniform: EXEC stays all-ones

  int m  = lane & 15;                         // M index within tile (A) / N index (B,C)
  int kh = lane >> 4;                         // half select: K pair {0,1} vs {2,3}

  const float* xrow = X + (wave * 16 + m) * K;
  const v2f* wpairs = (const v2f*)ldsW;       // [(k0>>1)*64 + n]

  v8f acc0 = 0, acc1 = 0, acc2 = 0, acc3 = 0;
#pragma unroll
  for (int kk = 0; kk < K; kk += 4) {
    int k0 = kk + 2 * kh;                     // even
    v2f A = *(const v2f*)(xrow + k0);         // global_load_b64, 8B aligned
    const v2f* wp = wpairs + (k0 >> 1) * 64 + m;
    v2f B0 = wp[0];                           // ds_load_b64 ×4
    v2f B1 = wp[16];
    v2f B2 = wp[32];
    v2f B3 = wp[48];
    acc0 = __builtin_amdgcn_wmma_f32_16x16x4_f32(false, A, false, B0, (short)0, acc0, false, false);
    acc1 = __builtin_amdgcn_wmma_f32_16x16x4_f32(false, A, false, B1, (short)0, acc1, false, false);
    acc2 = __builtin_amdgcn_wmma_f32_16x16x4_f32(false, A, false, B2, (short)0, acc2, false, false);
    acc3 = __builtin_amdgcn_wmma_f32_16x16x4_f32(false, A, false, B3, (short)0, acc3, false, false);
  }
  // C/D layout: VGPR r -> row (8*kh + r), col = m + 16*t
  float* ybase = Y + (wave * 16 + 8 * kh) * HIDC + m;
#pragma unroll
  for (int r = 0; r < 8; ++r) {
    ybase[r * HIDC +  0] = acc0[r];
    ybase[r * HIDC + 16] = acc1[r];
    ybase[r * HIDC + 32] = acc2[r];
    ybase[r * HIDC + 48] = acc3[r];
  }
}

// ---------- GCN scatter: acc[dst] += dinv[s]*dinv[d] * hw[src]  (16 thr/edge) ----------
__global__ void k_gcn_scatter(const float* __restrict__ hw, const int* __restrict__ src,
                              const int* __restrict__ dst, const float* __restrict__ dinv,
                              float* acc) {
  int t = blockIdx.x * blockDim.x + threadIdx.x;
  if (t >= ETOT * 16) return;
  int e = t >> 4, part = t & 15;
  int s, d;
  if (e < EREAL) { s = src[e]; d = dst[e]; } else { s = d = e - EREAL; }
  float norm = dinv[s] * dinv[d];
  const float4 v = *(const float4*)(hw + s * HIDC + part * 4);
  float* p = acc + d * HIDC + part * 4;
  atomicAdd(p + 0, norm * v.x);
  atomicAdd(p + 1, norm * v.y);
  atomicAdd(p + 2, norm * v.z);
  atomicAdd(p + 3, norm * v.w);
}

// ---------- bias + relu, in place ----------
__global__ void k_bias_relu(float* h, const float* __restrict__ b) {
  int i = blockIdx.x * blockDim.x + threadIdx.x;
  if (i < NN * HIDC) {
    float v = h[i] + b[i & 63];
    h[i] = v > 0.f ? v : 0.f;
  }
}

// ---------- GATv2 logits + segment max (1 thr per edge-head) ----------
__global__ void k_gat_logits(const float* __restrict__ hl, const float* __restrict__ hr,
                             const float* __restrict__ att, const int* __restrict__ src,
                             const int* __restrict__ dst, float* logits, unsigned* maxu) {
  int t = blockIdx.x * blockDim.x + threadIdx.x;
  if (t >= ETOT * 2) return;
  int e = t >> 1, h = t & 1;
  int s, d;
  if (e < EREAL) { s = src[e]; d = dst[e]; } else { s = d = e - EREAL; }
  const float* pl = hl + s * HIDC + h * 32;
  const float* pr = hr + d * HIDC + h * 32;
  const float* pa = att + h * 32;
  float a = 0.f;
#pragma unroll
  for (int c = 0; c < 32; ++c) {
    float v = pl[c] + pr[c];
    v = v > 0.f ? v : 0.2f * v;               // leaky relu 0.2
    a += pa[c] * v;
  }
  logits[t] = a;
  atomicMax(&maxu[d * 2 + h], ordered_u(a));
}

// ---------- exp(logit - max) + segment sum (overwrites logits with ex) ----------
__global__ void k_gat_exp(const int* __restrict__ dst, float* lg,
                          const unsigned* __restrict__ maxu, float* denom) {
  int t = blockIdx.x * blockDim.x + threadIdx.x;
  if (t >= ETOT * 2) return;
  int e = t >> 1, h = t & 1;
  int d = (e < EREAL) ? dst[e] : (e - EREAL);
  float m  = ordered_f(maxu[d * 2 + h]);      // every node has self-loop -> finite
  float ex = expf(lg[t] - m);
  lg[t] = ex;
  atomicAdd(&denom[d * 2 + h], ex);
}

// ---------- GAT aggregate: acc[dst] += alpha * hl[src]  (16 thr/edge) ----------
__global__ void k_gat_scatter(const float* __restrict__ hl, const float* __restrict__ ex,
                              const float* __restrict__ denom, const int* __restrict__ src,
                              const int* __restrict__ dst, float* acc) {
  int t = blockIdx.x * blockDim.x + threadIdx.x;
  if (t >= ETOT * 16) return;
  int e = t >> 4, part = t & 15, h = part >> 3;   // parts 0-7 -> head0, 8-15 -> head1
  int s, d;
  if (e < EREAL) { s = src[e]; d = dst[e]; } else { s = d = e - EREAL; }
  float alpha = ex[e * 2 + h] / (denom[d * 2 + h] + 1e-16f);
  const float4 v = *(const float4*)(hl + s * HIDC + part * 4);
  float* p = acc + d * HIDC + part * 4;
  atomicAdd(p + 0, alpha * v.x);
  atomicAdd(p + 1, alpha * v.y);
  atomicAdd(p + 2, alpha * v.z);
  atomicAdd(p + 3, alpha * v.w);
}

// ---------- final: relu(acc2 + b2) @ out_w + out_b  (1 thr per node) ----------
__global__ void k_out(const float* __restrict__ acc2, const float* __restrict__ b2,
                      const float* __restrict__ ow, const float* __restrict__ ob,
                      float* out) {
  int n = blockIdx.x * blockDim.x + threadIdx.x;
  if (n >= NN) return;
  const float* row = acc2 + n * HIDC;
  float a = 0.f;
#pragma unroll
  for (int c = 0; c < HIDC; ++c) {
    float v = row[c] + b2[c];
    v = v > 0.f ? v : 0.f;
    a += v * ow[c];
  }
  out[n] = a + ob[0];
}

extern "C" void kernel_launch(void* const* d_in, const int* in_sizes, int n_in,
                              void* d_out, int out_size, void* d_ws, size_t ws_size,
                              hipStream_t stream) {
  const float* x      = (const float*)d_in[0];
  const int*   eidx   = (const int*)d_in[1];
  const float* gcn1_w = (const float*)d_in[2];
  const float* gcn1_b = (const float*)d_in[3];
  const float* gat_wl = (const float*)d_in[4];
  const float* gat_wr = (const float*)d_in[5];
  const float* gat_at = (const float*)d_in[6];
  const float* gat_b  = (const float*)d_in[7];
  const float* gcn2_w = (const float*)d_in[8];
  const float* gcn2_b = (const float*)d_in[9];
  const float* out_w  = (const float*)d_in[10];
  const float* out_b  = (const float*)d_in[11];
  float* out = (float*)d_out;

  const int* src = eidx;            // edge_index[0]
  const int* dst = eidx + EREAL;    // edge_index[1]

  // workspace layout (bytes)
  char* ws = (char*)d_ws;
  const size_t NF = (size_t)NN * HIDC * sizeof(float);       // 25.6 MB
  float*    bufA   = (float*)(ws);                            // xw -> hl -> acc2
  float*    bufB   = (float*)(ws + NF);                       // acc1/h1 -> h2w
  float*    bufC   = (float*)(ws + 2 * NF);                   // hr -> gat acc/h2
  float*    logits = (float*)(ws + 3 * NF);                   // ETOT*2 floats (-> ex)
  float*    dinv   = (float*)(ws + 3 * NF + (size_t)ETOT * 2 * sizeof(float));
  unsigned* maxu   = (unsigned*)((char*)dinv + (size_t)NN * sizeof(float));
  float*    denom  = (float*)((char*)maxu + (size_t)NN * 2 * sizeof(unsigned));
  // total ~92.4 MB

  const int B = 256;
  dim3 blk(B);
  int gN    = (NN + B - 1) / B;
  int gE    = (EREAL + B - 1) / B;
  int gNF   = (NN * HIDC + B - 1) / B;
  int gE16  = (ETOT * 16 + B - 1) / B;
  int gE2   = (ETOT * 2 + B - 1) / B;
  int gGemm = (NN / 16 + (B / 32) - 1) / (B / 32);            // one wave per 16-row tile

  // zero accumulators
  hipMemsetAsync(bufB,  0, NF, stream);
  hipMemsetAsync(maxu,  0, (size_t)NN * 2 * sizeof(unsigned), stream);
  hipMemsetAsync(denom, 0, (size_t)NN * 2 * sizeof(float), stream);

  // degrees -> dinv
  k_init_deg<<<gN, blk, 0, stream>>>(dinv);
  k_degree<<<gE, blk, 0, stream>>>(dst, dinv);
  k_rsqrt<<<gN, blk, 0, stream>>>(dinv);

  // GCN1: xw = x @ W1 ; scatter ; bias+relu
  k_gemm_wmma<4><<<gGemm, blk, 0, stream>>>(x, gcn1_w, bufA);
  k_gcn_scatter<<<gE16, blk, 0, stream>>>(bufA, src, dst, dinv, bufB);
  k_bias_relu<<<gNF, blk, 0, stream>>>(bufB, gcn1_b);

  // GATv2: hl = h1@Wl (bufA), hr = h1@Wr (bufC)
  k_gemm_wmma<HIDC><<<gGemm, blk, 0, stream>>>(bufB, gat_wl, bufA);
  k_gemm_wmma<HIDC><<<gGemm, blk, 0, stream>>>(bufB, gat_wr, bufC);
  k_gat_logits<<<gE2, blk, 0, stream>>>(bufA, bufC, gat_at, src, dst, logits, maxu);
  hipMemsetAsync(bufC, 0, NF, stream);          // hr consumed; reuse as GAT accumulator
  k_gat_exp<<<gE2, blk, 0, stream>>>(dst, logits, maxu, denom);
  k_gat_scatter<<<gE16, blk, 0, stream>>>(bufA, logits, denom, src, dst, bufC);
  k_bias_relu<<<gNF, blk, 0, stream>>>(bufC, gat_b);

  // GCN2: h2w = h2 @ W2 (bufB) ; scatter into bufA ; fused bias+relu+out-proj
  k_gemm_wmma<HIDC><<<gGemm, blk, 0, stream>>>(bufC, gcn2_w, bufB);
  hipMemsetAsync(bufA, 0, NF, stream);          // hl consumed; reuse as accumulator
  k_gcn_scatter<<<gE16, blk, 0, stream>>>(bufB, src, dst, dinv, bufA);
  k_out<<<gN, blk, 0, stream>>>(bufA, gcn2_b, out_w, out_b, out);
}